// HarmonicOscillatorWithInteractionDold2_89137751261220
// MI455X (gfx1250) — compile-verified
//
#include <hip/hip_runtime.h>
#include <math.h>

typedef float v2f __attribute__((ext_vector_type(2)));
typedef float v8f __attribute__((ext_vector_type(8)));

#define NWALK 4096
#define NIN   24     // NPART*NDIM = 8*3
#define NINP  32     // padded input dim (rows 24..31 of W1 zeroed)
#define NH    256    // HIDDEN
#define LDW   257    // padded LDS row stride for [.][NH] arrays (bank-conflict-free)
#define LDZ   33     // padded LDS row stride for z tile

// Branch-free tanh: 1 - 2/(exp(2x)+1), via exp2 + fast rcp.
// Large |x| saturates through inf/0 without branches; fine in the
// non-saturated regime this network is initialized for.
__device__ __forceinline__ float fast_tanh(float x) {
    const float LOG2E_X2 = 2.8853900817779268f;   // 2*log2(e)
    float e = __builtin_amdgcn_exp2f(x * LOG2E_X2);
    float r = __builtin_amdgcn_rcpf(e + 1.0f);
    return 1.0f - 2.0f * r;
}

// One wave (32 threads) handles a tile of 16 walkers.
__global__ __launch_bounds__(32) void qmc_kinetic_wmma(
    const float* __restrict__ x,    // [4096][24]
    const float* __restrict__ W1,   // [24][256] row-major
    const float* __restrict__ b1,   // [256]
    const float* __restrict__ W2,   // [256]
    float* __restrict__ out)        // [4096]
{
    __shared__ float sW1[NINP * LDW]; // ~33 KB, padded rows 24..31 zero
    __shared__ float sS [NH];         // S_j = sum_i W1[i,j]^2
    __shared__ float sB1[NH];
    __shared__ float sW2[NH];
    __shared__ float sA [16 * LDW];   // a[walker][j], ~16.5 KB
    __shared__ float sZ [16 * LDZ];   // z tile [16][32+1]

    const int lane  = threadIdx.x;    // 0..31
    const int wtile = blockIdx.x;     // 0..255
    const int half  = lane >> 4;      // 0 or 1
    const int lm    = lane & 15;

    // ---- Stage weights into LDS (trip counts multiples of 32: no divergence) ----
    for (int idx = lane; idx < NINP * NH; idx += 32) {       // 8192
        int i = idx >> 8, j = idx & 255;
        sW1[i * LDW + j] = (i < NIN) ? W1[i * NH + j] : 0.0f;
    }
    for (int j = lane; j < NH; j += 32) {
        sB1[j] = b1[j];
        sW2[j] = W2[j];
    }
    // z tile, padded to 32 columns
    for (int idx = lane; idx < 16 * NINP; idx += 32) {       // 512
        int m = idx >> 5, i = idx & 31;
        sZ[m * LDZ + i] = (i < NIN) ? x[(wtile * 16 + m) * NIN + i] : 0.0f;
    }
    __syncthreads();

    // S_j (per-hidden-unit column norms of W1)
    for (int j = lane; j < NH; j += 32) {
        float s = 0.0f;
        #pragma unroll
        for (int i = 0; i < NIN; ++i) { float w = sW1[i * LDW + j]; s += w * w; }
        sS[j] = s;
    }
    __syncthreads();

    // ---- Phase 1: U = Z @ W1 + b1; h = tanh(u); a = (1-h^2)W2; laplacian accum ----
    float lap[8];
    #pragma unroll
    for (int r = 0; r < 8; ++r) lap[r] = 0.0f;

    for (int nt = 0; nt < 16; ++nt) {
        const int j    = lm + nt * 16;  // hidden index handled by this lane (N of tile)
        const float bj = sB1[j];
        v8f c;
        #pragma unroll
        for (int r = 0; r < 8; ++r) c[r] = bj;   // C init = b1 broadcast over M

        #pragma unroll
        for (int k = 0; k < 6; ++k) {            // K = 24 = 6 x 4
            const int kk = 4 * k + 2 * half;
            v2f av, bv;
            // A frag: lane L reg r = Z[M=L%16][K=2*(L/16)+r+4k]
            av[0] = sZ[lm * LDZ + kk + 0];
            av[1] = sZ[lm * LDZ + kk + 1];
            // B frag: lane L reg r = W1[K=2*(L/16)+r+4k][N=L%16+16nt]
            bv[0] = sW1[(kk + 0) * LDW + j];
            bv[1] = sW1[(kk + 1) * LDW + j];
            c = __builtin_amdgcn_wmma_f32_16x16x4_f32(
                    false, av, false, bv, (short)0, c, false, false);
        }

        const float w2j = sW2[j];
        const float Sj  = sS[j];
        #pragma unroll
        for (int r = 0; r < 8; ++r) {
            // c[r] = u for walker m = r + 8*half, hidden j
            float h    = fast_tanh(c[r]);
            float t    = 1.0f - h * h;
            float aval = t * w2j;
            lap[r] += (-2.0f * h * aval) * Sj;       // -2 h (1-h^2) W2_j * S_j
            sA[(r + 8 * half) * LDW + j] = aval;     // scatter a[walker][j]
        }
    }
    __syncthreads();

    // ---- Phase 2: G[16 x 24] = A[16 x 256] @ W1^T[256 x 24] (N padded to 32) ----
    v8f d0, d1;
    #pragma unroll
    for (int r = 0; r < 8; ++r) { d0[r] = 0.0f; d1[r] = 0.0f; }

    for (int k = 0; k < 64; ++k) {               // K = 256 = 64 x 4
        const int kk = 4 * k + 2 * half;
        v2f av, bv0, bv1;
        // A frag: a[walker=L%16][j=kk+r]
        av[0] = sA[lm * LDW + kk + 0];
        av[1] = sA[lm * LDW + kk + 1];
        // B frag: W1^T[K=j][N=i] = W1[i][j]; rows 24..31 are zero padding
        bv0[0] = sW1[lm * LDW + kk + 0];          // N-tile 0: i = 0..15
        bv0[1] = sW1[lm * LDW + kk + 1];
        bv1[0] = sW1[(lm + 16) * LDW + kk + 0];   // N-tile 1: i = 16..31
        bv1[1] = sW1[(lm + 16) * LDW + kk + 1];
        d0 = __builtin_amdgcn_wmma_f32_16x16x4_f32(
                 false, av, false, bv0, (short)0, d0, false, false);
        d1 = __builtin_amdgcn_wmma_f32_16x16x4_f32(
                 false, av, false, bv1, (short)0, d1, false, false);
    }

    // ---- Reduce: |g|^2 over N (16 lanes/half hold one dim-column each) + lap over j ----
    float tot[8];
    #pragma unroll
    for (int r = 0; r < 8; ++r)
        tot[r] = lap[r] + d0[r] * d0[r] + d1[r] * d1[r];

    #pragma unroll
    for (int msk = 1; msk < 16; msk <<= 1) {
        #pragma unroll
        for (int r = 0; r < 8; ++r)
            tot[r] += __shfl_xor(tot[r], msk, 32);   // masks < 16 stay within each half
    }

    if (lm == 0) {
        #pragma unroll
        for (int r = 0; r < 8; ++r) {
            int m = r + 8 * half;                    // walker within tile
            out[wtile * 16 + m] = -0.5f * tot[r];
        }
    }
}

extern "C" void kernel_launch(void* const* d_in, const int* in_sizes, int n_in,
                              void* d_out, int out_size, void* d_ws, size_t ws_size,
                              hipStream_t stream) {
    (void)in_sizes; (void)n_in; (void)d_ws; (void)ws_size;
    const float* x  = (const float*)d_in[0];   // [4096*24]
    const float* W1 = (const float*)d_in[1];   // [24*256]
    const float* b1 = (const float*)d_in[2];   // [256]
    const float* W2 = (const float*)d_in[3];   // [256]
    // d_in[4] = b2, unused (does not affect derivatives)
    float* out = (float*)d_out;                // [4096]

    dim3 grid(NWALK / 16), block(32);
    qmc_kinetic_wmma<<<grid, block, 0, stream>>>(x, W1, b1, W2, out);
}